// conv3x1x1_dcn_warp_16217796510039
// MI455X (gfx1250) — compile-verified
//
#include <hip/hip_runtime.h>
#include <hip/hip_bf16.h>

typedef float v2f __attribute__((ext_vector_type(2)));
typedef float v8f __attribute__((ext_vector_type(8)));

#define C_IN   256
#define OUT_C  256
#define T_DIM  16
#define H_DIM  32
#define W_DIM  32
#define HW     (H_DIM * W_DIM)        // 1024
#define THW    (T_DIM * HW)           // 16384
#define K_TOT  (3 * C_IN)             // 768
#define KCH    32                     // K chunk per iteration
#define KPAD   36                     // KCH + 4: 8B-aligned pairs, all-64-bank b64 loads,
                                      // and 256*KPAD floats = 2304 float4 = 9/thread
#define NCHUNK (K_TOT / KCH)          // 24
#define BCHUNK (OUT_C * KPAD)         // 9216 floats per prepacked weight chunk

// ---------------------------------------------------------------------------
// Kernel 0: prepack w_warp [O=256][K=768] into the padded LDS image
//   wTp[chunk][o][KPAD] : wTp[cb][o][kk] = w[o][cb*32+kk], pad kk=32..35 -> 0
// so the GEMM's B staging is a flat, fully coalesced float4 copy.
// ---------------------------------------------------------------------------
__global__ __launch_bounds__(256) void wprep_kernel(const float* __restrict__ w,
                                                    float* __restrict__ wTp) {
    const int cb = blockIdx.x;        // 0..23
    const int o  = threadIdx.x;       // 0..255
    float* dst = wTp + cb * BCHUNK + o * KPAD;
    const float* src = w + o * K_TOT + cb * KCH;
#pragma unroll
    for (int kk = 0; kk < KCH; ++kk) dst[kk] = src[kk];
#pragma unroll
    for (int kk = KCH; kk < KPAD; ++kk) dst[kk] = 0.f;
}

// ---------------------------------------------------------------------------
// Kernel 1: offset conv 3x3x3, 256 -> 4 channels, zero pad.
// One wave per (b, t, h) row; lane = w. 4 accumulators per lane.
// ---------------------------------------------------------------------------
__global__ __launch_bounds__(32) void offset_conv_kernel(const float* __restrict__ in,
                                                         const float* __restrict__ wOff,
                                                         float* __restrict__ offs) {
    const int bid = blockIdx.x;           // b*T*H + tt*H + h  (4*16*32 = 2048)
    const int h  = bid & 31;
    const int tt = (bid >> 5) & 15;
    const int b  = bid >> 9;
    const int wq = threadIdx.x;           // 0..31

    float acc0 = 0.f, acc1 = 0.f, acc2 = 0.f, acc3 = 0.f;

    for (int c = 0; c < C_IN; ++c) {
        float tap[27];
        int j = 0;
#pragma unroll
        for (int kz = 0; kz < 3; ++kz) {
            const int tz = tt - 1 + kz;
#pragma unroll
            for (int ky = 0; ky < 3; ++ky) {
                const int y = h - 1 + ky;
                const bool vzy = (tz >= 0) && (tz < T_DIM) && (y >= 0) && (y < H_DIM);
#pragma unroll
                for (int kx = 0; kx < 3; ++kx) {
                    const int x = wq - 1 + kx;
                    const bool v = vzy && (x >= 0) && (x < W_DIM);
                    const int idx = ((b * C_IN + c) * T_DIM + tz) * HW + y * W_DIM + x;
                    tap[j++] = v ? in[idx] : 0.f;
                }
            }
        }
        const float* wp = wOff + c * 27;   // wOff[o][c][27]
        float s0 = 0.f, s1 = 0.f, s2 = 0.f, s3 = 0.f;
#pragma unroll
        for (int q = 0; q < 27; ++q) {
            const float tv = tap[q];
            s0 += tv * wp[0 * C_IN * 27 + q];
            s1 += tv * wp[1 * C_IN * 27 + q];
            s2 += tv * wp[2 * C_IN * 27 + q];
            s3 += tv * wp[3 * C_IN * 27 + q];
        }
        acc0 += s0; acc1 += s1; acc2 += s2; acc3 += s3;
    }

    const int obase = ((b * 4 + 0) * T_DIM + tt) * HW + h * W_DIM + wq;
    offs[obase + 0 * THW] = acc0;
    offs[obase + 1 * THW] = acc1;
    offs[obase + 2 * THW] = acc2;
    offs[obase + 3 * THW] = acc3;
}

// ---------------------------------------------------------------------------
// Kernel 2: fused bilinear warp + 1x1x1 conv as WMMA GEMM.
//   M = B*T*H*W (tile 16/block; all 16 rows share b,tt,h -> same gather plane)
//   N = 256 outputs (16 tiles; 8 waves x 2 tiles each)
//   K = 768 = [fwd_feat | input | bwd_feat], chunked by 32.
// LDS layouts are k-contiguous+padded so every WMMA operand pair is one
// aligned ds_load_b64; B staging is a flat coalesced b128 copy.
// ---------------------------------------------------------------------------
__global__ __launch_bounds__(256) void fused_warp_gemm_kernel(const float* __restrict__ in,
                                                              const float* __restrict__ offs,
                                                              const float* __restrict__ wTp,
                                                              float* __restrict__ out) {
    __shared__ __align__(16) float As[16 * KPAD];   // 2.3 KB A tile (k contiguous)
    __shared__ __align__(16) float Bs[BCHUNK];      // 36 KB  B tile Bs[o][KPAD]
    __shared__ __align__(16) float Cs[16 * OUT_C];  // 16 KB  C tile Cs[row][o]
    __shared__ int   pixs[2][16][4];                // bilinear corner offsets in plane
    __shared__ float wts[2][16][4];                 // bilinear corner weights

    const int t = threadIdx.x;
    const int m_base = blockIdx.x * 16;
    const int w16 = m_base & 31;               // 0 or 16
    const int h   = (m_base >> 5) & 31;
    const int tt  = (m_base >> 10) & 15;
    const int b   = m_base >> 14;
    const int U   = b * (C_IN * THW) + tt * HW;  // plane base sans channel
    const int dpix = h * W_DIM + w16;            // direct-pixel base for row 0

    // ---- per-row bilinear metadata (threads 0..31: 16 rows x 2 directions)
    if (t < 32) {
        const int row = t & 15;
        const int dir = t >> 4;                  // 0 = fwd (ch 0,1), 1 = bwd (ch 2,3)
        const int wq  = w16 + row;
        const int ob  = ((b * 4 + 2 * dir) * T_DIM + tt) * HW + h * W_DIM + wq;
        const float dx = offs[ob];
        const float dy = offs[ob + THW];
        float gx = fminf(fmaxf(dx + (float)wq, 0.f), (float)(W_DIM - 1));
        float gy = fminf(fmaxf(dy + (float)h,  0.f), (float)(H_DIM - 1));
        const float x0f = floorf(gx), y0f = floorf(gy);
        const int x0 = (int)x0f, y0 = (int)y0f;
        const float wx = gx - x0f, wy = gy - y0f;
        const int x1 = (x0 + 1 < W_DIM) ? x0 + 1 : W_DIM - 1;
        const int y1 = (y0 + 1 < H_DIM) ? y0 + 1 : H_DIM - 1;
        pixs[dir][row][0] = y0 * W_DIM + x0;  wts[dir][row][0] = (1.f - wx) * (1.f - wy);
        pixs[dir][row][1] = y0 * W_DIM + x1;  wts[dir][row][1] = wx * (1.f - wy);
        pixs[dir][row][2] = y1 * W_DIM + x0;  wts[dir][row][2] = (1.f - wx) * wy;
        pixs[dir][row][3] = y1 * W_DIM + x1;  wts[dir][row][3] = wx * wy;
    }

    v8f acc0 = {0.f, 0.f, 0.f, 0.f, 0.f, 0.f, 0.f, 0.f};
    v8f acc1 = {0.f, 0.f, 0.f, 0.f, 0.f, 0.f, 0.f, 0.f};

    const int lane = t & 31;
    const int wv   = t >> 5;                 // wave id 0..7
    const int r    = lane & 15;
    const int sel  = (lane >> 4) << 1;       // lanes 0-15 -> K 0/1, lanes 16-31 -> K 2/3
    const int n0   = wv * 32;                // two 16-wide N tiles per wave

    for (int cb = 0; cb < NCHUNK; ++cb) {
        const int kb = cb * KCH;
        __syncthreads();   // previous chunk fully consumed before refill

        // ---- stage B chunk: flat coalesced b128 copy of the prepacked image
        {
            const float4* src = (const float4*)(wTp + cb * BCHUNK);
            float4*       dst = (float4*)Bs;
#pragma unroll
            for (int i = 0; i < 9; ++i)
                dst[i * 256 + t] = src[i * 256 + t];
        }

        // ---- build A chunk: 512 values, 2 per thread (warp thirds = 4-tap blend)
#pragma unroll
        for (int i = 0; i < 2; ++i) {
            const int v   = t + i * 256;
            const int row = v & 15;
            const int kk  = v >> 4;          // 0..31
            const int k   = kb + kk;
            const int third = k >> 8;        // chunk never straddles thirds (256 % 32 == 0)
            const int c     = k & 255;
            const float* plane = in + U + c * THW;
            float val;
            if (third == 1) {
                val = plane[dpix + row];
            } else {
                const int dir = third >> 1;  // 0 -> fwd, 1 -> bwd
                val = plane[pixs[dir][row][0]] * wts[dir][row][0]
                    + plane[pixs[dir][row][1]] * wts[dir][row][1]
                    + plane[pixs[dir][row][2]] * wts[dir][row][2]
                    + plane[pixs[dir][row][3]] * wts[dir][row][3];
            }
            As[row * KPAD + kk] = val;
        }
        __syncthreads();

        // hint the next weight chunk toward L2 while we do math
        if (cb + 1 < NCHUNK)
            __builtin_prefetch(wTp + (cb + 1) * BCHUNK + t * KPAD, 0, 1);

        // ---- 8 k-steps: 3 ds_load_b64 + 2 V_WMMA_F32_16X16X4_F32 per step
#pragma unroll
        for (int ks = 0; ks < 8; ++ks) {
            const int k0s = ks * 4 + sel;    // this lane-half's K pair
            v2f a  = *reinterpret_cast<const v2f*>(&As[r * KPAD + k0s]);
            v2f b0 = *reinterpret_cast<const v2f*>(&Bs[(n0 + r) * KPAD + k0s]);
            v2f b1 = *reinterpret_cast<const v2f*>(&Bs[(n0 + 16 + r) * KPAD + k0s]);
            acc0 = __builtin_amdgcn_wmma_f32_16x16x4_f32(false, a, false, b0,
                                                         (short)0, acc0, false, false);
            acc1 = __builtin_amdgcn_wmma_f32_16x16x4_f32(false, a, false, b1,
                                                         (short)0, acc1, false, false);
        }
    }

    __syncthreads();
    // ---- spill accumulators to LDS (D layout: VGPR r -> M = r / r+8, N = lane&15)
#pragma unroll
    for (int r8 = 0; r8 < 8; ++r8) {
        const int row = r8 + ((lane >> 4) << 3);
        Cs[row * OUT_C + n0 + r]      = acc0[r8];
        Cs[row * OUT_C + n0 + 16 + r] = acc1[r8];
    }
    __syncthreads();

    // ---- coalesced store: thread t = output channel o; 16 contiguous w per row
    float* dst = out + ((b * OUT_C + t) * T_DIM + tt) * HW + h * W_DIM + w16;
#pragma unroll
    for (int q = 0; q < 4; ++q) {
        float4 vv;
        vv.x = Cs[(q * 4 + 0) * OUT_C + t];
        vv.y = Cs[(q * 4 + 1) * OUT_C + t];
        vv.z = Cs[(q * 4 + 2) * OUT_C + t];
        vv.w = Cs[(q * 4 + 3) * OUT_C + t];
        ((float4*)dst)[q] = vv;
    }
}

// ---------------------------------------------------------------------------
extern "C" void kernel_launch(void* const* d_in, const int* in_sizes, int n_in,
                              void* d_out, int out_size, void* d_ws, size_t ws_size,
                              hipStream_t stream) {
    const float* in       = (const float*)d_in[0];   // [4,256,16,32,32]
    const float* w_offset = (const float*)d_in[1];   // [4,256,3,3,3]
    const float* w_warp   = (const float*)d_in[2];   // [256,768,1,1,1]
    float*       out      = (float*)d_out;           // [4,256,16,32,32]

    // workspace layout (floats): offsets [4,4,16,32,32] then wTp [24][256*36]
    float* offs = (float*)d_ws;                      // 262144 floats (1 MB)
    float* wTp  = offs + 4 * 4 * THW;                // 221184 floats (864 KB)

    wprep_kernel<<<NCHUNK, OUT_C, 0, stream>>>(w_warp, wTp);
    offset_conv_kernel<<<4 * T_DIM * H_DIM, 32, 0, stream>>>(in, w_offset, offs);

    const int m_tiles = (4 * T_DIM * HW) / 16;       // 4096
    fused_warp_gemm_kernel<<<m_tiles, 256, 0, stream>>>(in, offs, wTp, out);
}